// DCN_sep_2302102471256
// MI455X (gfx1250) — compile-verified
//
#include <hip/hip_runtime.h>

// ---------------------------------------------------------------------------
// Fused DCN_sep for MI455X (gfx1250, wave32).
//  Per block: one (b, h, w0..w0+15) output strip.
//   P1: stage feature im2col panel  SF[576][16] in LDS (K-pair interleaved)
//   P2: WMMA f32 16x16x4 GEMM  -> 216 offset/mask values/pos into LDS (sigmoid on mask)
//   P3: bilinear sample + modulate -> V[576][16] (reuses SF storage)
//   P4: WMMA f32 16x16x4 GEMM  -> 64 output channels -> global
// All matrix math in V_WMMA_F32_16X16X4_F32 to match the fp32 reference.
//
// LDS panel layout (sSF): element (k, n) at (k>>1)*32 + 2n + (k&1), so a
// B fragment {K=k, K=k+1} for even k is a contiguous, 8B-aligned ds_load_b64,
// and the two half-wave row groups span banks 0-31 / 32-63 (conflict-free).
// ---------------------------------------------------------------------------

typedef __attribute__((ext_vector_type(2))) float v2f;
typedef __attribute__((ext_vector_type(8))) float v8f;

#define CIN   64
#define HIMG  128
#define WIMG  128
#define KDIM  576      // CIN * 9
#define MOFF  144      // offset channels
#define MTOT  216      // offset + mask channels
#define NW    16       // strip width
#define OLD   17       // sO row stride (scalar access only)

#define SFIDX(k, n) ((((k) >> 1) << 5) + ((n) << 1) + ((k) & 1))

// Software-pipelined 16x16xKDIM f32 WMMA GEMM panel: A from global (one row
// per lane, b64 loads), B from LDS (one ds_load_b64 per step).
__device__ __forceinline__ v8f wmma_panel_f32(const float* __restrict__ wA,
                                              const float* __restrict__ sB,
                                              int koff, int col)
{
    const float* bp = sB + (col << 1);          // + (k+koff)*16 per step
    v8f acc = {};
    v2f a  = *(const v2f*)(wA + koff);
    v2f bb = *(const v2f*)(bp + koff * 16);
    #pragma unroll 4
    for (int k = 4; k < KDIM; k += 4) {
        v2f an = *(const v2f*)(wA + k + koff);
        v2f bn = *(const v2f*)(bp + (k + koff) * 16);
        acc = __builtin_amdgcn_wmma_f32_16x16x4_f32(
                  false, a, false, bb, (short)0, acc, false, false);
        a = an; bb = bn;
    }
    acc = __builtin_amdgcn_wmma_f32_16x16x4_f32(
              false, a, false, bb, (short)0, acc, false, false);
    return acc;
}

__global__ __launch_bounds__(128) void dcn_fused_kernel(
    const float* __restrict__ inp,   const float* __restrict__ feat,
    const float* __restrict__ w_off, const float* __restrict__ b_off,
    const float* __restrict__ w_msk, const float* __restrict__ b_msk,
    const float* __restrict__ w_dcn, const float* __restrict__ b_dcn,
    float* __restrict__ out)
{
    __shared__ float sSF[KDIM * NW];    // feature im2col panel; reused as sampled panel V
    __shared__ float sO [MTOT * OLD];   // offset (0..143) + mask (144..215) panel

    const int tid  = threadIdx.x;
    const int lane = tid & 31;
    const int wv   = tid >> 5;          // wave id 0..3
    const int col  = lane & 15;         // N within 16-wide tile
    const int half = lane >> 4;         // 0: K=k,k+1   1: K=k+2,k+3
    const int koff = half * 2;

    const int blk = blockIdx.x;         // 4096 blocks = 4 * 128 * 8
    const int w0  = (blk & 7) * NW;
    const int h   = (blk >> 3) & 127;
    const int b   = blk >> 10;

    // ---- Phase 1: feature im2col panel -> LDS -------------------------------
    {
        const float* fB = feat + (size_t)b * CIN * HIMG * WIMG;
        for (int e = tid; e < KDIM * NW; e += 128) {          // 9216 = 72 * 128
            int n   = e & 15;
            int row = e >> 4;                                  // cin*9 + kk
            int ci  = row / 9, kk = row % 9;
            int y   = h  + kk / 3 - 1;
            int x   = w0 + n + (kk % 3) - 1;
            float v = 0.0f;
            if (y >= 0 && y < HIMG && x >= 0 && x < WIMG)
                v = fB[(ci * HIMG + y) * WIMG + x];
            sSF[SFIDX(row, n)] = v;
        }
    }
    __syncthreads();

    // ---- Phase 2: offset+mask implicit GEMM (216 x 576 x 16) ----------------
    for (int p = 0; p < 4; ++p) {
        int mt = p * 4 + wv;            // 14 M-tiles of 16 (tile 13 partial)
        if (mt >= 14) continue;         // wave-uniform: EXEC stays all-ones for WMMA
        int mrow = mt * 16 + col;       // A row this lane supplies
        // Lanes with mrow >= MTOT read a dummy valid row (w_off row 0): the
        // resulting D rows are discarded below, and D row m depends only on
        // A row m, so no valid output is polluted. No per-lane predication.
        const float* wA;
        if      (mrow < MOFF) wA = w_off + (size_t)mrow * KDIM;
        else if (mrow < MTOT) wA = w_msk + (size_t)(mrow - MOFF) * KDIM;
        else                  wA = w_off;
        v8f acc = wmma_panel_f32(wA, sSF, koff, col);
        #pragma unroll
        for (int i = 0; i < 8; ++i) {
            int m = mt * 16 + i + half * 8;
            if (m < MTOT) {
                float v;
                if (m < MOFF) {
                    v = acc[i] + b_off[m];
                } else {
                    v = acc[i] + b_msk[m - MOFF];
                    v = 1.0f / (1.0f + __expf(-v));           // sigmoid for mask
                }
                sO[m * OLD + col] = v;
            }
        }
    }
    __syncthreads();   // phase-2 LDS reads done; safe to overwrite sSF

    // ---- Phase 3: bilinear sample + modulate -> V panel (reuse sSF) ---------
    {
        const float* inpB = inp + (size_t)b * CIN * HIMG * WIMG;
        for (int e = tid; e < 8 * 9 * NW; e += 128) {          // 1152 = 9 * 128
            int n  = e & 15;
            int r  = e >> 4;            // dg*9 + kk
            int kk = r % 9;
            int dg = r / 9;
            int orow = (dg * 9 + kk) * 2;
            float dy = sO[ orow      * OLD + n];
            float dx = sO[(orow + 1) * OLD + n];
            float mk = sO[(MOFF + dg * 9 + kk) * OLD + n];
            float ys = dy + (float)(h  + kk / 3 - 1);
            float xs = dx + (float)(w0 + n + (kk % 3) - 1);
            float y0f = floorf(ys), x0f = floorf(xs);
            float wy = ys - y0f, wx = xs - x0f;
            int y0 = (int)y0f, x0 = (int)x0f;
            int y1 = y0 + 1,   x1 = x0 + 1;
            float vy0 = (y0 >= 0 && y0 < HIMG) ? 1.0f : 0.0f;
            float vy1 = (y1 >= 0 && y1 < HIMG) ? 1.0f : 0.0f;
            float vx0 = (x0 >= 0 && x0 < WIMG) ? 1.0f : 0.0f;
            float vx1 = (x1 >= 0 && x1 < WIMG) ? 1.0f : 0.0f;
            int cy0 = min(max(y0, 0), HIMG - 1), cy1 = min(max(y1, 0), HIMG - 1);
            int cx0 = min(max(x0, 0), WIMG - 1), cx1 = min(max(x1, 0), WIMG - 1);
            int i00 = cy0 * WIMG + cx0, i01 = cy0 * WIMG + cx1;
            int i10 = cy1 * WIMG + cx0, i11 = cy1 * WIMG + cx1;
            float q00 = (1.0f - wy) * (1.0f - wx) * vy0 * vx0;
            float q01 = (1.0f - wy) * wx          * vy0 * vx1;
            float q10 = wy          * (1.0f - wx) * vy1 * vx0;
            float q11 = wy          * wx          * vy1 * vx1;
            #pragma unroll
            for (int cg = 0; cg < 8; ++cg) {
                int c = dg * 8 + cg;
                const float* p = inpB + (size_t)c * HIMG * WIMG;
                float v = q00 * p[i00] + q01 * p[i01] + q10 * p[i10] + q11 * p[i11];
                sSF[SFIDX(c * 9 + kk, n)] = mk * v;            // row = c*9+kk matches w_dcn flat
            }
        }
    }
    __syncthreads();

    // ---- Phase 4: DCN GEMM (64 x 576 x 16), one M-tile per wave -------------
    {
        int mrow = wv * 16 + col;
        const float* wA = w_dcn + (size_t)mrow * KDIM;
        v8f acc = wmma_panel_f32(wA, sSF, koff, col);
        #pragma unroll
        for (int i = 0; i < 8; ++i) {
            int m = wv * 16 + i + half * 8;
            out[(((size_t)b * 64 + m) * HIMG + h) * WIMG + w0 + col] = acc[i] + b_dcn[m];
        }
    }
}

extern "C" void kernel_launch(void* const* d_in, const int* in_sizes, int n_in,
                              void* d_out, int out_size, void* d_ws, size_t ws_size,
                              hipStream_t stream) {
    const float* inp   = (const float*)d_in[0];
    const float* feat  = (const float*)d_in[1];
    const float* w_off = (const float*)d_in[2];
    const float* b_off = (const float*)d_in[3];
    const float* w_msk = (const float*)d_in[4];
    const float* b_msk = (const float*)d_in[5];
    const float* w_dcn = (const float*)d_in[6];
    const float* b_dcn = (const float*)d_in[7];
    (void)in_sizes; (void)n_in; (void)out_size; (void)d_ws; (void)ws_size;
    dcn_fused_kernel<<<4096, 128, 0, stream>>>(
        inp, feat, w_off, b_off, w_msk, b_msk, w_dcn, b_dcn, (float*)d_out);
}